// HetNet_60052232733109
// MI455X (gfx1250) — compile-verified
//
#include <hip/hip_runtime.h>

typedef __attribute__((ext_vector_type(2))) float v2f;
typedef __attribute__((ext_vector_type(8))) float v8f;

#define AS 34          // LDS act tile row stride (floats); bank-conflict-free for frag loads/stores
#define WAVES 4
#define TPB (WAVES * 32)

// Single-instruction ReLU: v_max_num_f32 dst, 0, src (skips fmaxf's canonicalize).
__device__ __forceinline__ float relu_f(float x) {
  float r;
  asm("v_max_num_f32 %0, 0, %1" : "=v"(r) : "v"(x));
  return r;
}

// One 32x32 fp32 layer on a 16-point LDS tile via V_WMMA_F32_16X16X4_F32.
// act: [16][AS] activations (row m = point, col k 0..31). W row-major [k][n].
// b0/b1 = per-lane bias for the two N-halves (preloaded by caller, loop-invariant).
__device__ __forceinline__ void frag_layer(const float* act, const float* __restrict__ W,
                                           float b0, float b1, v8f& c0, v8f& c1) {
  const int lane = threadIdx.x & 31;
  const int hi = lane >> 4;      // which K half this lane serves
  const int lo = lane & 15;
  c0 = (v8f){b0, b0, b0, b0, b0, b0, b0, b0};
  c1 = (v8f){b1, b1, b1, b1, b1, b1, b1, b1};
#pragma unroll
  for (int ks = 0; ks < 8; ++ks) {
    const int k0 = ks * 4 + 2 * hi;
    // A (16x4 f32): lanes 0-15 rows, K=k0..k0+1; lanes 16-31 same rows, K=k0+2..k0+3
    v2f a;  a.x  = act[lo * AS + k0];        a.y  = act[lo * AS + k0 + 1];
    // B (4x16 f32): VGPR0 = rows K=k0 (lanes 0-15) / K=k0+2 (lanes 16-31), VGPR1 = K=k0+1 / k0+3
    v2f bA; bA.x = W[k0 * 32 + lo];          bA.y = W[(k0 + 1) * 32 + lo];          // N-half 0
    v2f bB; bB.x = W[k0 * 32 + 16 + lo];     bB.y = W[(k0 + 1) * 32 + 16 + lo];     // N-half 1
    c0 = __builtin_amdgcn_wmma_f32_16x16x4_f32(false, a, false, bA, (short)0, c0, false, false);
    c1 = __builtin_amdgcn_wmma_f32_16x16x4_f32(false, a, false, bB, (short)0, c1, false, false);
  }
}

__device__ __forceinline__ void store_tile(float* act, v8f c0, v8f c1, bool relu) {
  const int lane = threadIdx.x & 31;
  const int hi = lane >> 4;
  const int lo = lane & 15;
#pragma unroll
  for (int v = 0; v < 8; ++v) {
    const int m = v + 8 * hi;
    float x0 = c0[v], x1 = c1[v];
    if (relu) { x0 = relu_f(x0); x1 = relu_f(x1); }
    act[m * AS + lo]      = x0;
    act[m * AS + 16 + lo] = x1;
  }
}

// relu(raw*w0 + L) -> [32x32 relu] -> [32x32 relu] -> mean over P points per group.
// raw index = b*sb + u*su + p*sp ; L row index = b*tb + u*tu + p*tp (L has row stride 32).
__global__ __launch_bounds__(TPB)
void k_rank1_mlp2_mean(const float* __restrict__ raw, int sb, int su, int sp,
                       const float* __restrict__ L, int tb, int tu, int tp,
                       const float* __restrict__ w0,
                       const float* __restrict__ W2, const float* __restrict__ b2,
                       const float* __restrict__ W3, const float* __restrict__ b3,
                       float* __restrict__ out, int U, int P, float invP) {
  __shared__ float act[WAVES][16 * AS];
  __shared__ float part[WAVES][32];
  const int wave = threadIdx.x >> 5;
  const int lane = threadIdx.x & 31;
  const int hi = lane >> 4, lo = lane & 15;
  const int b = blockIdx.x / U;
  const int u = blockIdx.x - b * U;
  const float w0l = w0[lane];
  const float b2l0 = b2[lo], b2l1 = b2[16 + lo];   // hoisted bias loads
  const float b3l0 = b3[lo], b3l1 = b3[16 + lo];
  float* a = act[wave];
  float s0 = 0.f, s1 = 0.f;
  const int nT = (P + 15) >> 4;
  for (int t = wave; t < nT; t += WAVES) {
    const int pbase = t * 16;
#pragma unroll 4
    for (int m = 0; m < 16; ++m) {
      int p = pbase + m; if (p > P - 1) p = P - 1;   // clamp padded rows (masked at sum)
      const float r  = raw[b * sb + u * su + p * sp];
      const float lv = L[(b * tb + u * tu + p * tp) * 32 + lane];
      a[m * AS + lane] = relu_f(fmaf(r, w0l, lv));
    }
    v8f c0, c1;
    frag_layer(a, W2, b2l0, b2l1, c0, c1);
    store_tile(a, c0, c1, true);
    frag_layer(a, W3, b3l0, b3l1, c0, c1);
#pragma unroll
    for (int v = 0; v < 8; ++v) {
      const int p = pbase + v + 8 * hi;
      if (p < P) { s0 += relu_f(c0[v]); s1 += relu_f(c1[v]); }
    }
  }
  s0 += __shfl_xor(s0, 16, 32);   // combine row halves per column
  s1 += __shfl_xor(s1, 16, 32);
  if (lane < 16) { part[wave][lo] = s0; part[wave][16 + lo] = s1; }
  __syncthreads();
  if (threadIdx.x < 32) {
    const float t0 = part[0][lane] + part[1][lane] + part[2][lane] + part[3][lane];
    out[(b * U + u) * 32 + lane] = t0 * invP;
  }
}

// 1..3 layer 32x32 chain on [N,32] points; optional elementwise-add second input.
// Intermediate layers always ReLU; final layer ReLU iff relu_last. W2/W3 null => fewer layers.
__global__ __launch_bounds__(TPB)
void k_mlp_chain(const float* __restrict__ in0, const float* __restrict__ in1,
                 const float* __restrict__ W1, const float* __restrict__ b1,
                 const float* __restrict__ W2, const float* __restrict__ b2,
                 const float* __restrict__ W3, const float* __restrict__ b3,
                 int relu_last, float* __restrict__ out, int N) {
  __shared__ float act[WAVES][16 * AS];
  const int wave = threadIdx.x >> 5, lane = threadIdx.x & 31;
  const int lo = lane & 15;
  const int tile = blockIdx.x * WAVES + wave;
  if (tile * 16 >= N) return;
  float* a = act[wave];
  const int r0 = tile * 16;
#pragma unroll 4
  for (int m = 0; m < 16; ++m) {
    float x = in0[(r0 + m) * 32 + lane];
    if (in1) x += in1[(r0 + m) * 32 + lane];
    a[m * AS + lane] = x;
  }
  v8f c0, c1;
  frag_layer(a, W1, b1[lo], b1[16 + lo], c0, c1);
  const bool last1 = (W2 == nullptr);
  store_tile(a, c0, c1, last1 ? (relu_last != 0) : true);
  if (!last1) {
    frag_layer(a, W2, b2[lo], b2[16 + lo], c0, c1);
    const bool last2 = (W3 == nullptr);
    store_tile(a, c0, c1, last2 ? (relu_last != 0) : true);
    if (!last2) {
      frag_layer(a, W3, b3[lo], b3[16 + lo], c0, c1);
      store_tile(a, c0, c1, relu_last != 0);
    }
  }
#pragma unroll 4
  for (int m = 0; m < 16; ++m)
    out[(r0 + m) * 32 + lane] = a[m * AS + lane];
}

// Final head: relu(Pz[b,q]+Py[b,j]) -> 32x32 relu (WMMA) -> dot with w3 + b3 -> out[b,q,j].
__global__ __launch_bounds__(TPB)
void k_fy_out(const float* __restrict__ Pz, const float* __restrict__ Py,
              const float* __restrict__ W2, const float* __restrict__ b2,
              const float* __restrict__ w3, const float* __restrict__ b3,
              float* __restrict__ out, int NqJ, int J, int N) {
  __shared__ float act[WAVES][16 * AS];
  const int wave = threadIdx.x >> 5, lane = threadIdx.x & 31;
  const int lo = lane & 15;
  const int tile = blockIdx.x * WAVES + wave;
  if (tile * 16 >= N) return;
  float* a = act[wave];
  const int p0 = tile * 16;
#pragma unroll 4
  for (int m = 0; m < 16; ++m) {
    const int pt = p0 + m;
    const int bq = pt / J;                         // b*Nq + q
    const int bj = (pt / NqJ) * J + (pt - bq * J); // b*J + j
    a[m * AS + lane] = relu_f(Pz[bq * 32 + lane] + Py[bj * 32 + lane]);
  }
  v8f c0, c1;
  frag_layer(a, W2, b2[lo], b2[16 + lo], c0, c1);
  store_tile(a, c0, c1, true);
  if (lane < 16) {
    float s = b3[0];
#pragma unroll
    for (int n = 0; n < 32; ++n) s = fmaf(a[lane * AS + n], w3[n], s);
    out[p0 + lane] = s;
  }
}

extern "C" void kernel_launch(void* const* d_in, const int* in_sizes, int n_in,
                              void* d_out, int out_size, void* d_ws, size_t ws_size,
                              hipStream_t stream) {
  (void)n_in; (void)out_size; (void)ws_size;
  const int B = 32, Ns = 512, Nq = 512, F = 64, J = 8;
  const float* que_x = (const float*)d_in[0];
  const float* sup_x = (const float*)d_in[1];
  const float* sup_y = (const float*)d_in[2];

  // chains: v,c,uf,ug,fv,gv,fz,gz,fy ; each = [W1,b1,W2,b2,W3,b3]
  enum { V = 0, C, UF, UG, FV, GV, FZ, GZ, FY };
  int pos[9];
  if (in_sizes[3] == 32) {           // insertion order (dense_v first: W(1,32)=32)
    for (int i = 0; i < 9; ++i) pos[i] = i;
  } else {                           // JAX sorted-key order: c,fv,fy,fz,gv,gz,uf,ug,v
    pos[V] = 8; pos[C] = 0; pos[UF] = 6; pos[UG] = 7; pos[FV] = 1;
    pos[GV] = 4; pos[FZ] = 3; pos[GZ] = 5; pos[FY] = 2;
  }
  const float* Pm[9][6];
  for (int c = 0; c < 9; ++c)
    for (int k = 0; k < 6; ++k)
      Pm[c][k] = (const float*)d_in[3 + 6 * pos[c] + k];

  float* ws = (float*)d_ws;
  float* tmp_vx = ws;              // [B*F,32]  65536
  float* tmp_vy = ws + 65536;      // [B*J,32]   8192
  float* vs     = ws + 73728;      // 65536
  float* cs     = ws + 139264;     //  8192
  float* Lx     = ws + 147456;     // 65536  vs@W_uf1[1:]+b
  float* Ly     = ws + 212992;     //  8192
  float* mfvx   = ws + 221184;     // 65536
  float* mfvy   = ws + 286720;     //  8192
  float* in_xs  = ws + 294912;     // 65536
  float* in_ys  = ws + 360448;     //  8192
  float* L_fz   = ws + 368640;     // 65536
  float* Py     = ws + 434176;     //  8192
  float* S1     = ws + 442368;     // 3 big reusable slots of 524288 floats
  float* S2     = S1 + 524288;
  float* S3     = S2 + 524288;
  float* zbias  = S3 + 524288;     // 32 zero floats (bias for the bias-less Py linear)
  float* u_xs = S1, * u_ys = S2, * u_s = S3;
  float* L_fv = S1, * mean_fz = S2, * z = S3, * Pz = S1;

  hipMemsetAsync(zbias, 0, 32 * sizeof(float), stream);   // capturable memset node

  dim3 blk(TPB);
  // vs/cs encoders: dense_v (rank-1 head, bias-only "latent") + mean over Ns
  k_rank1_mlp2_mean<<<B * F, blk, 0, stream>>>(sup_x, Ns * F, 1, F, Pm[V][1], 0, 0, 0,
      Pm[V][0], Pm[V][2], Pm[V][3], Pm[V][4], Pm[V][5], tmp_vx, F, Ns, 1.f / Ns);
  k_rank1_mlp2_mean<<<B * J, blk, 0, stream>>>(sup_y, Ns * J, 1, J, Pm[V][1], 0, 0, 0,
      Pm[V][0], Pm[V][2], Pm[V][3], Pm[V][4], Pm[V][5], tmp_vy, J, Ns, 1.f / Ns);
  k_mlp_chain<<<32, blk, 0, stream>>>(tmp_vx, nullptr, Pm[C][0], Pm[C][1], Pm[C][2], Pm[C][3],
      Pm[C][4], Pm[C][5], 1, vs, B * F);
  k_mlp_chain<<<4, blk, 0, stream>>>(tmp_vy, nullptr, Pm[C][0], Pm[C][1], Pm[C][2], Pm[C][3],
      Pm[C][4], Pm[C][5], 1, cs, B * J);
  // latent precompute for dense_uf layer 1 (rows 1..32 of the (33,32) weight)
  k_mlp_chain<<<32, blk, 0, stream>>>(vs, nullptr, Pm[UF][0] + 32, Pm[UF][1], nullptr, nullptr,
      nullptr, nullptr, 0, Lx, B * F);
  k_mlp_chain<<<4, blk, 0, stream>>>(cs, nullptr, Pm[UF][0] + 32, Pm[UF][1], nullptr, nullptr,
      nullptr, nullptr, 0, Ly, B * J);
  // u_xs / u_ys: mean over F (resp. J) per (b,s)
  k_rank1_mlp2_mean<<<B * Ns, blk, 0, stream>>>(sup_x, Ns * F, F, 1, Lx, F, 0, 1,
      Pm[UF][0], Pm[UF][2], Pm[UF][3], Pm[UF][4], Pm[UF][5], u_xs, Ns, F, 1.f / F);
  k_rank1_mlp2_mean<<<B * Ns, blk, 0, stream>>>(sup_y, Ns * J, J, 1, Ly, J, 0, 1,
      Pm[UF][0], Pm[UF][2], Pm[UF][3], Pm[UF][4], Pm[UF][5], u_ys, Ns, J, 1.f / J);
  // u_s = dense_ug(u_xs + u_ys)
  k_mlp_chain<<<256, blk, 0, stream>>>(u_xs, u_ys, Pm[UG][0], Pm[UG][1], Pm[UG][2], Pm[UG][3],
      Pm[UG][4], Pm[UG][5], 1, u_s, B * Ns);
  // dense_fv latent: L_fv = u_s @ W_fv1[1:] + b_fv1
  k_mlp_chain<<<256, blk, 0, stream>>>(u_s, nullptr, Pm[FV][0] + 32, Pm[FV][1], nullptr, nullptr,
      nullptr, nullptr, 0, L_fv, B * Ns);
  // mean over Ns per (b,f) / (b,j)
  k_rank1_mlp2_mean<<<B * F, blk, 0, stream>>>(sup_x, Ns * F, 1, F, L_fv, Ns, 0, 1,
      Pm[FV][0], Pm[FV][2], Pm[FV][3], Pm[FV][4], Pm[FV][5], mfvx, F, Ns, 1.f / Ns);
  k_rank1_mlp2_mean<<<B * J, blk, 0, stream>>>(sup_y, Ns * J, 1, J, L_fv, Ns, 0, 1,
      Pm[FV][0], Pm[FV][2], Pm[FV][3], Pm[FV][4], Pm[FV][5], mfvy, J, Ns, 1.f / Ns);
  k_mlp_chain<<<32, blk, 0, stream>>>(mfvx, nullptr, Pm[GV][0], Pm[GV][1], Pm[GV][2], Pm[GV][3],
      Pm[GV][4], Pm[GV][5], 1, in_xs, B * F);
  k_mlp_chain<<<4, blk, 0, stream>>>(mfvy, nullptr, Pm[GV][0], Pm[GV][1], Pm[GV][2], Pm[GV][3],
      Pm[GV][4], Pm[GV][5], 1, in_ys, B * J);
  // dense_fz latent + query path: mean over F per (b,q)
  k_mlp_chain<<<32, blk, 0, stream>>>(in_xs, nullptr, Pm[FZ][0] + 32, Pm[FZ][1], nullptr, nullptr,
      nullptr, nullptr, 0, L_fz, B * F);
  k_rank1_mlp2_mean<<<B * Nq, blk, 0, stream>>>(que_x, Nq * F, F, 1, L_fz, F, 0, 1,
      Pm[FZ][0], Pm[FZ][2], Pm[FZ][3], Pm[FZ][4], Pm[FZ][5], mean_fz, Nq, F, 1.f / F);
  k_mlp_chain<<<256, blk, 0, stream>>>(mean_fz, nullptr, Pm[GZ][0], Pm[GZ][1], Pm[GZ][2], Pm[GZ][3],
      Pm[GZ][4], Pm[GZ][5], 1, z, B * Nq);
  // dense_fy layer-1 split: Pz = z@W[0:32]+b ; Py = in_ys@W[32:64] (zero bias)
  k_mlp_chain<<<256, blk, 0, stream>>>(z, nullptr, Pm[FY][0], Pm[FY][1], nullptr, nullptr,
      nullptr, nullptr, 0, Pz, B * Nq);
  k_mlp_chain<<<4, blk, 0, stream>>>(in_ys, nullptr, Pm[FY][0] + 1024, zbias, nullptr, nullptr,
      nullptr, nullptr, 0, Py, B * J);
  k_fy_out<<<(B * Nq * J) / 64, blk, 0, stream>>>(Pz, Py, Pm[FY][2], Pm[FY][3],
      Pm[FY][4], Pm[FY][5], (float*)d_out, Nq * J, J, B * Nq * J);
}